// LinearAttentionConditioner_71459665871283
// MI455X (gfx1250) — compile-verified
//
#include <hip/hip_runtime.h>
#include <cmath>

// Problem constants (from reference): N=32, C=64, H=32, W=32, A=4
#define SEQ      65536              // C*H*W
#define NBATCH   32
#define CH       2048               // positions per chunk (one wave per chunk in pass 3)
#define NCH      (SEQ / CH)         // 32 chunks per batch  (== wave32 lanes, used in pass 2)
#define NCHUNKS  (NBATCH * NCH)     // 1024 chunks total
#define TPC      (CH / 16)          // 128 16-position tiles per chunk
#define STATE    20                 // 16 (k⊗v) + 4 (z)
#define VXS      20                 // padded row stride (floats) of [V|1|0..] in LDS (16B aligned)

typedef float v2f __attribute__((ext_vector_type(2)));
typedef float v8f __attribute__((ext_vector_type(8)));

__device__ __forceinline__ float softplus1(float x) {
  // stable log(1+e^x), matches jax.nn.softplus
  return fmaxf(x, 0.f) + log1pf(expf(-fabsf(x)));
}

static __device__ __forceinline__ v8f wmma_16x16x4_f32(v2f a, v2f b, v8f c) {
  // D(16x16,f32) = A(16x4,f32) x B(4x16,f32) + C ; K=4 == head dim A of this model
  return __builtin_amdgcn_wmma_f32_16x16x4_f32(false, a, false, b, (short)0, c, false, false);
}

// ---------------------------------------------------------------------------
// Pass 1: per-chunk partial sums of (k ⊗ v, k). One 256-thread block per chunk.
// ---------------------------------------------------------------------------
__global__ __launch_bounds__(256) void la_pass1_chunksums(
    const float* __restrict__ x, const float* __restrict__ in_w,
    const float* __restrict__ in_b, float* __restrict__ chunk_sums) {
  const int chunk = blockIdx.x;
  const int n  = chunk / NCH;
  const int s0 = (chunk % NCH) * CH;

  const float4* WK = reinterpret_cast<const float4*>(in_w) + SEQ;      // p=1 (k)
  const float4* WV = reinterpret_cast<const float4*>(in_w) + 2 * SEQ;  // p=2 (v)
  const float4* BK = reinterpret_cast<const float4*>(in_b) + SEQ;
  const float4* BV = reinterpret_cast<const float4*>(in_b) + 2 * SEQ;

  float st[STATE];
#pragma unroll
  for (int i = 0; i < STATE; ++i) st[i] = 0.f;

  for (int t = threadIdx.x; t < CH; t += 256) {
    const int s = s0 + t;
    const float xv = x[n * SEQ + s];
    const float4 wk = WK[s], bk = BK[s], wv = WV[s], bv = BV[s];
    const float kk[4] = {softplus1(fmaf(wk.x, xv, bk.x)), softplus1(fmaf(wk.y, xv, bk.y)),
                         softplus1(fmaf(wk.z, xv, bk.z)), softplus1(fmaf(wk.w, xv, bk.w))};
    const float vv[4] = {fmaf(wv.x, xv, bv.x), fmaf(wv.y, xv, bv.y),
                         fmaf(wv.z, xv, bv.z), fmaf(wv.w, xv, bv.w)};
#pragma unroll
    for (int i = 0; i < 4; ++i) {
#pragma unroll
      for (int j = 0; j < 4; ++j) st[i * 4 + j] = fmaf(kk[i], vv[j], st[i * 4 + j]);
      st[16 + i] += kk[i];
    }
  }

  // wave32 tree reduce, then cross-wave via LDS
#pragma unroll
  for (int i = 0; i < STATE; ++i) {
#pragma unroll
    for (int off = 16; off > 0; off >>= 1) st[i] += __shfl_down(st[i], off, 32);
  }
  __shared__ float red[8][STATE];
  const int wave = threadIdx.x >> 5, lane = threadIdx.x & 31;
  if (lane == 0) {
#pragma unroll
    for (int i = 0; i < STATE; ++i) red[wave][i] = st[i];
  }
  __syncthreads();
  if (threadIdx.x < STATE) {
    float s = 0.f;
#pragma unroll
    for (int w = 0; w < 8; ++w) s += red[w][threadIdx.x];
    chunk_sums[chunk * STATE + threadIdx.x] = s;
  }
}

// ---------------------------------------------------------------------------
// Pass 2: exclusive scan over the NCH(=32) chunk states of each batch.
// One wave32 per batch; lane == chunk index, shfl_up inclusive scan.
// ---------------------------------------------------------------------------
__global__ __launch_bounds__(32) void la_pass2_scan(
    const float* __restrict__ chunk_sums, float* __restrict__ prefix) {
  const int n = blockIdx.x, lane = threadIdx.x;
  float v[STATE];
#pragma unroll
  for (int i = 0; i < STATE; ++i) v[i] = chunk_sums[(n * NCH + lane) * STATE + i];
#pragma unroll
  for (int i = 0; i < STATE; ++i) {
    for (int off = 1; off < 32; off <<= 1) {
      const float t = __shfl_up(v[i], off, 32);
      if (lane >= off) v[i] += t;
    }
  }
#pragma unroll
  for (int i = 0; i < STATE; ++i) {
    const float e = __shfl_up(v[i], 1, 32);
    prefix[(n * NCH + lane) * STATE + i] = (lane == 0) ? 0.f : e;
  }
}

// ---------------------------------------------------------------------------
// Pass 3: chunked causal linear attention via v_wmma_f32_16x16x4_f32.
// One wave per 2048-position chunk; 4 waves per block. Per 16-pos tile:
//   scores = Q x K^T                (1 WMMA, K=4 = head dim)
//   acc    = Q x [S | z]            (1 WMMA, cross-tile numer + denom)
//   acc   += mask(scores) x [V | 1] (4 chained WMMAs, intra-tile numer+denom)
// A/B operand components are loaded per lane-half as float2 (no register-array
// selects); the padded [V|1|0..] matrix keeps all LDS reads unconditional.
// ---------------------------------------------------------------------------
__global__ __launch_bounds__(128) void la_pass3_scan_attn(
    const float* __restrict__ x, const float* __restrict__ in_w,
    const float* __restrict__ in_b, const float* __restrict__ hid_w,
    const float* __restrict__ hid_b, const float* __restrict__ out_w,
    const float* __restrict__ out_b, const float* __restrict__ prefix,
    float* __restrict__ out) {
  const int wave = threadIdx.x >> 5, lane = threadIdx.x & 31;
  const int chunk = blockIdx.x * 4 + wave;
  const int n  = chunk / NCH;
  const int s0 = (chunk % NCH) * CH;
  const int col  = lane & 15;               // D/B column, also this lane's tile position
  const int half = lane >> 4;               // 0: K rows {0,1}, 1: K rows {2,3}
  const int c0   = half * 2;
  const int mrow = half * 8;                // D row offset of this lane's half

  __shared__ float lds_sc[4][16][17];       // masked scores, padded stride
  __shared__ float lds_k[4][16][4];         // k rows (16B/row, float4 store)
  __shared__ float lds_vx[4][16][VXS];      // [V | 1 | 0..] padded to 16 cols, stride 20
  __shared__ float lds_y[4][16][5];         // numer[4] + denom per position

  const float2* W2 = reinterpret_cast<const float2*>(in_w);   // component pairs
  const float2* B2 = reinterpret_cast<const float2*>(in_b);
  const float4* WQ = reinterpret_cast<const float4*>(in_w);
  const float4* BQ = reinterpret_cast<const float4*>(in_b);
  const float4* WV = WQ + 2 * SEQ;
  const float4* BV = BQ + 2 * SEQ;
  const float4* HW = reinterpret_cast<const float4*>(hid_w);
  const float*  pref = prefix + chunk * STATE;

  // one-time init of the constant columns of [V|1|0..]: col 4 = 1, cols 5..15 = 0
  if (lane < 16) {
#pragma unroll
    for (int j = 4; j < 16; ++j) lds_vx[wave][lane][j] = (j == 4) ? 1.f : 0.f;
  }

  // Carry state [S | z | 0...] directly in B-operand layout:
  // lane holds B[c0][col] and B[c0+1][col].
  v2f sB;
  sB.x = (col < 4) ? pref[c0 * 4 + col]       : ((col == 4) ? pref[16 + c0]     : 0.f);
  sB.y = (col < 4) ? pref[(c0 + 1) * 4 + col] : ((col == 4) ? pref[16 + c0 + 1] : 0.f);

  // y at position s0-1 (for the right-shift); ones at the sequence start.
  float yc0 = 1.f, yc1 = 1.f, yc2 = 1.f, yc3 = 1.f;
  if (s0 != 0) {
    const int sm = s0 - 1;
    const float xv = x[n * SEQ + sm];
    const float4 wq = WQ[sm], bq = BQ[sm];
    const float q[4] = {softplus1(fmaf(wq.x, xv, bq.x)), softplus1(fmaf(wq.y, xv, bq.y)),
                        softplus1(fmaf(wq.z, xv, bq.z)), softplus1(fmaf(wq.w, xv, bq.w))};
    float num[4] = {0.f, 0.f, 0.f, 0.f};
    float den = 0.f;
#pragma unroll
    for (int i = 0; i < 4; ++i) {
      den = fmaf(q[i], pref[16 + i], den);
#pragma unroll
      for (int j = 0; j < 4; ++j) num[j] = fmaf(q[i], pref[i * 4 + j], num[j]);
    }
    const float inv = 1.f / den;
    yc0 = num[0] * inv; yc1 = num[1] * inv; yc2 = num[2] * inv; yc3 = num[3] * inv;
  }

  for (int tt = 0; tt < TPC; ++tt) {
    const int p0  = s0 + tt * 16;
    const int pos = p0 + col;

    // prefetch next tile's streams into cache (global_prefetch_b8)
    if (tt + 1 < TPC) {
      __builtin_prefetch(&x[n * SEQ + pos + 16], 0, 0);
      __builtin_prefetch(&WQ[pos + 16], 0, 0);
      __builtin_prefetch(&HW[pos + 16], 0, 0);
    }

    const float xv = x[n * SEQ + pos];

    // A = Q (16x4): lane holds row `col`, K pair {c0,c0+1} -> one float2 load
    const float2 wq2 = W2[2 * pos + half];
    const float2 bq2 = B2[2 * pos + half];
    v2f aq;
    aq.x = softplus1(fmaf(wq2.x, xv, bq2.x));
    aq.y = softplus1(fmaf(wq2.y, xv, bq2.y));

    // B = K^T (4x16): same per-lane-half pattern (k plane p=1)
    const float2 wk2 = W2[2 * (SEQ + pos) + half];
    const float2 bk2 = B2[2 * (SEQ + pos) + half];
    v2f bkv;
    bkv.x = softplus1(fmaf(wk2.x, xv, bk2.x));
    bkv.y = softplus1(fmaf(wk2.y, xv, bk2.y));

    // lanes<16 own row `col`: complete k (high half only) + v, store to LDS
    if (lane < 16) {
      const float2 wkh = W2[2 * (SEQ + pos) + 1];
      const float2 bkh = B2[2 * (SEQ + pos) + 1];
      const float k2 = softplus1(fmaf(wkh.x, xv, bkh.x));
      const float k3 = softplus1(fmaf(wkh.y, xv, bkh.y));
      const float4 wv = WV[pos], bv = BV[pos];
      const float v0 = fmaf(wv.x, xv, bv.x), v1 = fmaf(wv.y, xv, bv.y);
      const float v2 = fmaf(wv.z, xv, bv.z), v3 = fmaf(wv.w, xv, bv.w);
      *reinterpret_cast<float4*>(&lds_k[wave][col][0])  = make_float4(bkv.x, bkv.y, k2, k3);
      *reinterpret_cast<float4*>(&lds_vx[wave][col][0]) = make_float4(v0, v1, v2, v3);
    }

    v8f zero8 = {0.f, 0.f, 0.f, 0.f, 0.f, 0.f, 0.f, 0.f};

    v8f sc = wmma_16x16x4_f32(aq, bkv, zero8);          // scores[s][t] = q[s].k[t]
#pragma unroll
    for (int r = 0; r < 8; ++r) {                       // causal mask: keep t <= s
      const int M = r + mrow;
      sc[r] = (col <= M) ? sc[r] : 0.f;
    }

    v8f acc = wmma_16x16x4_f32(aq, sB, zero8);          // cross part: Q x [S|z]

#pragma unroll
    for (int r = 0; r < 8; ++r) lds_sc[wave][r + mrow][col] = sc[r];
    asm volatile("s_wait_dscnt 0" ::: "memory");        // D-layout -> A-layout transpose

#pragma unroll
    for (int tb = 0; tb < 4; ++tb) {                    // += mask(scores) x [V|1]
      v2f am;
      am.x = lds_sc[wave][col][4 * tb + c0];
      am.y = lds_sc[wave][col][4 * tb + c0 + 1];
      const int r0 = 4 * tb + c0;
      v2f bvv;                                          // unconditional padded loads
      bvv.x = lds_vx[wave][r0][col];
      bvv.y = lds_vx[wave][r0 + 1][col];
      acc = wmma_16x16x4_f32(am, bvv, acc);
    }

    if (col < 5) {                                      // numer(0..3), denom(4) per position
#pragma unroll
      for (int r = 0; r < 8; ++r) lds_y[wave][r + mrow][col] = acc[r];
    }
    asm volatile("s_wait_dscnt 0" ::: "memory");

    // advance carried state: sB += K^T x [V|1] over this tile (B layout, scalar FMAs)
    {
      float d0 = 0.f, d1 = 0.f;
#pragma unroll
      for (int t = 0; t < 16; ++t) {
        const float kk0 = lds_k[wave][t][c0];
        const float kk1 = lds_k[wave][t][c0 + 1];
        const float vvt = lds_vx[wave][t][col];         // unconditional padded load
        d0 = fmaf(kk0, vvt, d0);
        d1 = fmaf(kk1, vvt, d1);
      }
      sB.x += d0; sB.y += d1;
    }

    // shifted output: position p0+L consumes y[p0+L-1]; then per-position projections
    if (lane < 16) {
      const int ip = (lane + 15) & 15;                  // lane-1 (lane 0 -> dummy row 15)
      const float inv = 1.f / lds_y[wave][ip][4];
      float y0 = lds_y[wave][ip][0] * inv;
      float y1 = lds_y[wave][ip][1] * inv;
      float y2 = lds_y[wave][ip][2] * inv;
      float y3 = lds_y[wave][ip][3] * inv;
      y0 = (lane == 0) ? yc0 : y0;                      // plain v_cndmask selects
      y1 = (lane == 0) ? yc1 : y1;
      y2 = (lane == 0) ? yc2 : y2;
      y3 = (lane == 0) ? yc3 : y3;
      const int p = p0 + lane;
      float o = out_b[p];
#pragma unroll
      for (int ao = 0; ao < 4; ++ao) {
        const float4 hw = HW[ao * SEQ + p];
        float h = hid_b[ao * SEQ + p];
        h = fmaf(hw.x, y0, h); h = fmaf(hw.y, y1, h);
        h = fmaf(hw.z, y2, h); h = fmaf(hw.w, y3, h);
        o = fmaf(tanhf(h), out_w[ao * SEQ + p], o);
      }
      out[n * SEQ + p] = o;
    }

    // new carry = y at the tile's last position (uniform LDS broadcast read)
    {
      const float inv = 1.f / lds_y[wave][15][4];
      yc0 = lds_y[wave][15][0] * inv;
      yc1 = lds_y[wave][15][1] * inv;
      yc2 = lds_y[wave][15][2] * inv;
      yc3 = lds_y[wave][15][3] * inv;
    }
  }
}

// ---------------------------------------------------------------------------
extern "C" void kernel_launch(void* const* d_in, const int* in_sizes, int n_in,
                              void* d_out, int out_size, void* d_ws, size_t ws_size,
                              hipStream_t stream) {
  (void)in_sizes; (void)n_in; (void)out_size; (void)ws_size;
  const float* x     = (const float*)d_in[0];
  const float* in_w  = (const float*)d_in[1];
  const float* in_b  = (const float*)d_in[2];
  const float* hid_w = (const float*)d_in[3];
  const float* hid_b = (const float*)d_in[4];
  const float* out_w = (const float*)d_in[5];
  const float* out_b = (const float*)d_in[6];
  float* out = (float*)d_out;

  float* chunk_sums = (float*)d_ws;                       // NCHUNKS * 20 floats
  float* prefix     = chunk_sums + NCHUNKS * STATE;       // NCHUNKS * 20 floats

  la_pass1_chunksums<<<NCHUNKS, 256, 0, stream>>>(x, in_w, in_b, chunk_sums);
  la_pass2_scan<<<NBATCH, 32, 0, stream>>>(chunk_sums, prefix);
  la_pass3_scan_attn<<<NCHUNKS / 4, 128, 0, stream>>>(x, in_w, in_b, hid_w, hid_b,
                                                      out_w, out_b, prefix, out);
}